// TransposedConvTF_28664611734012
// MI455X (gfx1250) — compile-verified
//
#include <hip/hip_runtime.h>
#include <stdint.h>

typedef __attribute__((ext_vector_type(2))) float v2f;
typedef __attribute__((ext_vector_type(8))) float v8f;

namespace {
constexpr int Bn   = 8;
constexpr int Hn   = 256;
constexpr int Wn   = 256;
constexpr int CI   = 64;
constexpr int CO   = 64;
constexpr int OHn  = 258;
constexpr int OWn  = 258;
constexpr int QT   = 32;                  // output columns per block
constexpr int NQ   = (OWn + QT - 1) / QT; // 9
constexpr int PT   = 2;                   // output rows per block (B-frag reuse)
constexpr int NP   = OHn / PT;            // 129
constexpr int INC  = QT + 2;              // 34 staged input columns (halo 2)
constexpr int ROWS = PT + 2;              // 4 staged input rows (halo 2)
constexpr int NPIX = ROWS * INC;          // 136
constexpr int CS   = 68;                  // channel stride (dwords): b64 across wave hits 4n+j -> all 64 banks once
}

__global__ __launch_bounds__(256)
void tconv_f32_wmma(const float* __restrict__ in,
                    const float* __restrict__ wt,
                    float* __restrict__ out)
{
    __shared__ float sIn[NPIX * CS];   // input tile [pix][ci], 36992 B, zero-padded halo
    __shared__ float sW [CO   * CS];   // one weight tap [co][ci], 17408 B

    const int tid  = threadIdx.x;
    int bidx = blockIdx.x;
    const int q0 = (bidx % NQ) * QT;  bidx /= NQ;
    const int p  = (bidx % NP) * PT;  bidx /= NP;
    const int b  =  bidx;

    // ---- stage input tile: rows p-2..p+1, cols q0-2..q0+31, zero outside bounds ----
    const float* inB = in + (size_t)b * Hn * Wn * CI;
    for (int idx = tid; idx < NPIX * CI; idx += 256) {
        const int pix = idx >> 6;          // 0..135
        const int ci  = idx & 63;          // consecutive lanes -> consecutive ci: coalesced global read
        const int r   = pix / INC;
        const int c   = pix - r * INC;
        const int y   = p  - 2 + r;
        const int x   = q0 - 2 + c;
        float v = 0.0f;
        if ((unsigned)y < (unsigned)Hn && (unsigned)x < (unsigned)Wn)
            v = inB[((size_t)(y * Wn + x)) * CI + ci];
        sIn[pix * CS + ci] = v;
    }

    const int lane = tid & 31;
    const int wave = tid >> 5;
    const int mT   = wave >> 2;            // 0..1 : 16-column group
    const int nT   = wave & 3;             // 0..3 : 16-cout group
    const int l16  = lane & 15;
    const int koff = (lane >> 4) << 1;     // 0 | 2 : WMMA f32 K split per lane half

    v8f acc0 = {};                         // output row p
    v8f acc1 = {};                         // output row p+1

    for (int tap = 0; tap < 9; ++tap) {
        const int dy = tap / 3;
        const int dx = tap - dy * 3;

        __syncthreads();                   // prior-tap reads of sW complete
        // ---- stage flipped tap sW[co][ci] = wt[ci][co][2-dy][2-dx] via async global->LDS ----
        {
            const int tapoff = (2 - dy) * 3 + (2 - dx);
            const float* wTap = wt + tapoff;
            for (int idx = tid; idx < CO * CI; idx += 256) {
                const int ci = idx >> 6;
                const int co = idx & 63;
                const uint64_t gaddr = (uint64_t)(wTap + ((size_t)(ci * CO + co)) * 9);
                const uint32_t laddr = (uint32_t)(uintptr_t)&sW[co * CS + ci];
                asm volatile("global_load_async_to_lds_b32 %0, %1, off"
                             :: "v"(laddr), "v"(gaddr) : "memory");
            }
            asm volatile("s_wait_asynccnt 0x0" ::: "memory");
        }
        __syncthreads();

        // A row g (g=0,1 -> out rows p,p+1) lives at staged row dy+g; B shared across rows.
        const float* aP0 = &sIn[((dy    ) * INC + mT * 16 + l16 + dx) * CS + koff];
        const float* aP1 = &sIn[((dy + 1) * INC + mT * 16 + l16 + dx) * CS + koff];
        const float* bP  = &sW [(nT * 16 + l16) * CS + koff];
        #pragma unroll
        for (int k0 = 0; k0 < CI; k0 += 4) {
            const v2f bFrag = *(const v2f*)(bP  + k0);  // ds_load_b64, conflict-free
            const v2f a0    = *(const v2f*)(aP0 + k0);
            const v2f a1    = *(const v2f*)(aP1 + k0);
            acc0 = __builtin_amdgcn_wmma_f32_16x16x4_f32(
                       false, a0, false, bFrag, (short)0, acc0, false, false);
            acc1 = __builtin_amdgcn_wmma_f32_16x16x4_f32(
                       false, a1, false, bFrag, (short)0, acc1, false, false);
        }
    }

    // ---- store C tiles: VGPR v -> M = v + 8*(lane>=16), N = lane&15 ----
    float* o0 = out + ((size_t)b * OHn + p) * OWn * CO;   // p+1 <= 257 always (OH even)
    float* o1 = o0 + (size_t)OWn * CO;
    const int co  = nT * 16 + l16;
    const int mHi = (lane >> 4) << 3;
    #pragma unroll
    for (int v = 0; v < 8; ++v) {
        const int q = q0 + mT * 16 + mHi + v;
        if (q < OWn) {
            o0[(size_t)q * CO + co] = acc0[v];
            o1[(size_t)q * CO + co] = acc1[v];
        }
    }
}

extern "C" void kernel_launch(void* const* d_in, const int* in_sizes, int n_in,
                              void* d_out, int out_size, void* d_ws, size_t ws_size,
                              hipStream_t stream)
{
    const float* in  = (const float*)d_in[0];   // [8,256,256,64] f32
    const float* wt  = (const float*)d_in[1];   // [64,64,3,3]    f32
    float*       out = (float*)d_out;           // [8,258,258,64] f32

    dim3 grid(NQ * NP * Bn);                    // 9 * 129 * 8 = 9288 blocks
    tconv_f32_wmma<<<grid, 256, 0, stream>>>(in, wt, out);
}